// Model_13322988552561
// MI455X (gfx1250) — compile-verified
//
#include <hip/hip_runtime.h>
#include <math.h>

typedef float v2f __attribute__((ext_vector_type(2)));
typedef float v8f __attribute__((ext_vector_type(8)));
typedef unsigned int u32x4 __attribute__((ext_vector_type(4)));
typedef int i32x8 __attribute__((ext_vector_type(8)));
typedef int i32x4 __attribute__((ext_vector_type(4)));

#define NB    8
#define TIN   96
#define TOUTP 12
#define NN    512
#define RCH   32
#define CLH   64
#define NBN   (NB*NN)          // 4096
#define SEQ32 (RCH*TIN)        // 3072
#define SEQ64 (CLH*TIN)        // 6144
#define EPSV  1e-5f

static __device__ __forceinline__ v8f wmma4(v2f a, v2f b, v8f c) {
  // D = A(16x4) * B(4x16) + C, fp32 — emits v_wmma_f32_16x16x4_f32
  return __builtin_amdgcn_wmma_f32_16x16x4_f32(false, a, false, b, (short)0, c, false, false);
}

// ---------------- input 1x1 conv over C_IN=2 (temporal RC=32 / spatial CL=64) ----
__global__ void in_conv_kernel(const float* __restrict__ hist, const float* __restrict__ w,
                               const float* __restrict__ bias, float* __restrict__ out, int CO)
{
  int idx = blockIdx.x*blockDim.x + threadIdx.x;      // over NBN*TIN
  if (idx >= NBN*TIN) return;
  int t = idx % TIN; int bn = idx / TIN;
  int b = bn / NN, n = bn % NN;
  const float* xp = hist + ((size_t)(b*TIN + t)*NN + n)*2;
  float x0 = xp[0], x1 = xp[1];
  float* ob = out + (size_t)bn*CO*TIN;
  for (int c=0;c<CO;++c) ob[c*TIN + t] = w[c*2]*x0 + w[c*2+1]*x1 + bias[c];
}

// ---------------- WaveNet gated dilated conv, WMMA f32 16x16x4 + TDM staging -----
template<int KK>
__global__ __launch_bounds__(32) void gated_conv_wmma(
    const float* __restrict__ res, float* __restrict__ gf,
    const float* __restrict__ fw, const float* __restrict__ fb,
    const float* __restrict__ gw, const float* __restrict__ gb,
    int Lin, int dil)
{
  __shared__ float sX [RCH][TIN];
  __shared__ float sWf[RCH][TIN];
  __shared__ float sWg[RCH][TIN];
  const int bn = blockIdx.x;
  const int lane = threadIdx.x;
  constexpr int K = RCH*KK;                   // 64 or 96 (compile-time)

  // --- Tensor Data Mover: DMA the whole 32x96 fp32 sequence tile into LDS -------
  {
    unsigned long long ga = (unsigned long long)(uintptr_t)(res + (size_t)bn*SEQ32);
    unsigned lds = (unsigned)(uintptr_t)(&sX[0][0]);       // generic LDS addr[31:0] == LDS offset
    // D# group0: count=1 | lds_addr | global_addr(57b) | type=2
    u32x4 g0 = { 1u, lds, (unsigned)(ga & 0xFFFFFFFFu),
                 (unsigned)((ga >> 32) & 0x01FFFFFFu) | (2u << 30) };
    // D# group1: data_size=4B; tensor_dim0=tile_dim0=stride0=3072; tensor_dim1=tile_dim1=1
    i32x8 g1 = { (int)(2u << 16),                 // data_size=2 (4 bytes)
                 (int)((unsigned)(SEQ32 & 0xFFFF) << 16),   // tensor_dim0[15:0]
                 (int)(((unsigned)SEQ32 >> 16) | (1u << 16)),// tensor_dim0[31:16] | tensor_dim1[15:0]
                 (int)((unsigned)(SEQ32 & 0xFFFF) << 16),   // tile_dim0
                 1,                                          // tile_dim1=1, tile_dim2=0
                 (int)SEQ32,                                 // tensor_dim0_stride[31:0]
                 0, 0 };
    i32x4 gz4 = { 0, 0, 0, 0 };
    i32x8 gz8 = { 0, 0, 0, 0, 0, 0, 0, 0 };
    __builtin_amdgcn_tensor_load_to_lds(g0, g1, gz4, gz4, gz8, 0);
  }
  // weights staged by the wave while the TDM runs
  #pragma unroll
  for (int q=0;q<K;++q) { sWf[lane][q] = fw[lane*K+q]; sWg[lane][q] = gw[lane*K+q]; }
  __builtin_amdgcn_s_wait_tensorcnt(0);
  __syncthreads();

  const int Lout = Lin - dil*(KK-1);
  const int m  = lane & 15;
  const int hi = lane >> 4;
  float* ob = gf + (size_t)bn*SEQ32;
  for (int mt=0; mt<2; ++mt) {
    const int arow = mt*16 + m;
    for (int nt=0; nt*16 < Lout; ++nt) {
      v8f accF = {0,0,0,0,0,0,0,0};
      v8f accG = {0,0,0,0,0,0,0,0};
      const int col = nt*16 + m;
      const int cc  = (col < Lout) ? col : (Lout-1);   // clamp, garbage cols discarded
      #pragma unroll
      for (int k0=0;k0<K;k0+=4) {
        const int kl = k0 + 2*hi;
        v2f aF = { sWf[arow][kl], sWf[arow][kl+1] };
        v2f aG = { sWg[arow][kl], sWg[arow][kl+1] };
        v2f bv = { sX[kl/KK][cc + dil*(kl%KK)], sX[(kl+1)/KK][cc + dil*((kl+1)%KK)] };
        accF = wmma4(aF, bv, accF);
        accG = wmma4(aG, bv, accG);
      }
      #pragma unroll
      for (int r=0;r<8;++r) {
        const int co = mt*16 + r + 8*hi;
        const int t  = nt*16 + m;
        if (t < Lout) {
          float f = accF[r] + fb[co];
          float g = accG[r] + gb[co];
          ob[co*TIN + t] = tanhf(f) * (1.0f/(1.0f+__expf(-g)));
        }
      }
    }
  }
}

// ---------------- 1x1 conv (32x32) + optional shifted add (skip / residual) ------
__global__ void pw32_shift_add(const float* __restrict__ gf, const float* __restrict__ addsrc,
                               float* __restrict__ dst, const float* __restrict__ w,
                               const float* __restrict__ bias, int Lout, int shift, int doAdd)
{
  size_t idx = (size_t)blockIdx.x*blockDim.x + threadIdx.x;
  size_t total = (size_t)NBN*RCH*Lout;
  if (idx >= total) return;
  int t = (int)(idx % Lout); size_t r = idx / Lout;
  int c = (int)(r % RCH); int bn = (int)(r / RCH);
  const float* g = gf + (size_t)bn*SEQ32;
  float acc = bias[c];
  for (int cc=0; cc<RCH; ++cc) acc += w[c*RCH+cc]*g[cc*TIN + t];
  if (doAdd) acc += addsrc[(size_t)bn*SEQ32 + c*TIN + t + shift];
  dst[(size_t)bn*SEQ32 + c*TIN + t] = acc;
}

// ---------------- ConvTranspose (full-pad conv with flipped kernel) --------------
__global__ void tconv_kernel(const float* __restrict__ in, float* __restrict__ out,
                             const float* __restrict__ tw, const float* __restrict__ tb,
                             int Lin, int dil, int kk)
{
  const int pad = dil*(kk-1);
  const int LoutT = Lin + pad;
  size_t idx = (size_t)blockIdx.x*blockDim.x + threadIdx.x;
  size_t total = (size_t)NBN*RCH*LoutT;
  if (idx >= total) return;
  int t = (int)(idx % LoutT); size_t r = idx / LoutT;
  int c = (int)(r % RCH); int bn = (int)(r / RCH);
  const float* ip = in + (size_t)bn*SEQ32;
  float acc = tb[c];
  for (int ci=0; ci<RCH; ++ci)
    for (int j=0;j<kk;++j) {
      int ti = t - pad + j*dil;
      if (ti>=0 && ti<Lin) acc += tw[(ci*RCH + c)*kk + (kk-1-j)] * ip[ci*TIN + ti];
    }
  out[(size_t)bn*SEQ32 + c*TIN + t] = acc;
}

// ---------------- Linear over time: (12,96) --------------------------------------
__global__ void lin_time_kernel(const float* __restrict__ skip, float* __restrict__ outs,
                                const float* __restrict__ lw, const float* __restrict__ lb)
{
  size_t idx = (size_t)blockIdx.x*blockDim.x + threadIdx.x;
  size_t total = (size_t)NBN*RCH*TOUTP;
  if (idx >= total) return;
  int p = (int)(idx % TOUTP); size_t r = idx / TOUTP;
  int c = (int)(r % RCH); int bn = (int)(r / RCH);
  const float* sp = skip + (size_t)bn*SEQ32 + c*TIN;
  float acc = lb[p];
  for (int t=0;t<TIN;++t) acc += lw[p*TIN+t]*sp[t];
  outs[((size_t)bn*RCH + c)*TOUTP + p] = acc;
}

// ---------------- temporal stack combine + final ---------------------------------
__global__ void temp_final_kernel(const float* __restrict__ outs, float* __restrict__ tout,
                                  const float* __restrict__ sw2, const float* __restrict__ sb2,
                                  const float* __restrict__ fw, const float* __restrict__ fb)
{
  size_t idx = (size_t)blockIdx.x*blockDim.x + threadIdx.x;
  if (idx >= (size_t)NB*TOUTP*NN) return;
  int n = (int)(idx % NN); size_t r = idx / NN;
  int p = (int)(r % TOUTP); int b = (int)(r / TOUTP);
  int bn = b*NN + n;
  size_t stride = (size_t)NBN*RCH*TOUTP;
  float acc = fb[0];
  for (int c=0;c<RCH;++c) {
    float v = sb2[0];
    v += outs[((size_t)bn*RCH+c)*TOUTP+p]*sw2[0];
    v += outs[stride + ((size_t)bn*RCH+c)*TOUTP+p]*sw2[1];
    acc += fw[c]*v;
  }
  tout[((size_t)b*TOUTP+p)*NN + n] = acc;
}

// ---------------- head K/Q/V convs (valid, kernel ks over time) ------------------
__global__ void kqv_kernel(const float* __restrict__ HS,
                           const float* __restrict__ kw, const float* __restrict__ kb,
                           const float* __restrict__ qw, const float* __restrict__ qb,
                           const float* __restrict__ vw, const float* __restrict__ vb,
                           float* __restrict__ K, float* __restrict__ Q, float* __restrict__ V,
                           int ks)
{
  const int Tp = TIN - ks + 1;
  size_t idx = (size_t)blockIdx.x*blockDim.x + threadIdx.x;
  if (idx >= (size_t)NBN*TIN) return;
  int c = (int)(idx % TIN); int bn = (int)(idx / TIN);
  int b = bn / NN, n = bn % NN;
  float aK[TOUTP], aQ[TOUTP], aV[TOUTP];
  int valid = (c < Tp);
  for (int to=0;to<TOUTP;++to){
    aK[to] = valid ? kb[to] : 0.f;
    aQ[to] = valid ? qb[to] : 0.f;
    aV[to] = valid ? vb[to] : 0.f;
  }
  if (valid) {
    const float* hp = HS + (size_t)bn*SEQ64;
    for (int cl=0;cl<CLH;++cl)
      for (int dt=0;dt<ks;++dt) {
        float x = hp[cl*TIN + c + dt];
        #pragma unroll
        for (int to=0;to<TOUTP;++to) {
          int wi = (to*CLH+cl)*ks+dt;
          aK[to] += kw[wi]*x; aQ[to] += qw[wi]*x; aV[to] += vw[wi]*x;
        }
      }
  }
  for (int to=0;to<TOUTP;++to) {
    size_t o = ((size_t)(b*TOUTP+to)*NN + n)*TIN + c;
    K[o]=aK[to]; Q[o]=aQ[to]; V[o]=aV[to];        // zero-padded tail -> exact K=96 WMMA chain
  }
}

// ---------------- S = K * Q^T  (32x32 tile per wave, WMMA) -----------------------
__global__ __launch_bounds__(32) void attn_s_wmma(const float* __restrict__ K,
                                                  const float* __restrict__ Q,
                                                  float* __restrict__ S, int b)
{
  const int to = blockIdx.z;
  const int n0 = blockIdx.x*32, m0 = blockIdx.y*32;
  const float* Kp = K + (size_t)(b*TOUTP+to)*NN*TIN;
  const float* Qp = Q + (size_t)(b*TOUTP+to)*NN*TIN;
  const int lane = threadIdx.x, m = lane & 15, hi = lane >> 4;
  v8f a00={0,0,0,0,0,0,0,0}, a01=a00, a10=a00, a11=a00;
  #pragma unroll
  for (int k0=0;k0<TIN;k0+=4) {
    int kl = k0 + 2*hi;
    v2f ar0 = { Kp[(n0+m)*TIN+kl],    Kp[(n0+m)*TIN+kl+1]    };
    v2f ar1 = { Kp[(n0+16+m)*TIN+kl], Kp[(n0+16+m)*TIN+kl+1] };
    v2f bc0 = { Qp[(m0+m)*TIN+kl],    Qp[(m0+m)*TIN+kl+1]    };
    v2f bc1 = { Qp[(m0+16+m)*TIN+kl], Qp[(m0+16+m)*TIN+kl+1] };
    a00 = wmma4(ar0,bc0,a00); a01 = wmma4(ar0,bc1,a01);
    a10 = wmma4(ar1,bc0,a10); a11 = wmma4(ar1,bc1,a11);
  }
  float* Sp = S + (size_t)to*NN*NN;
  #pragma unroll
  for (int r=0;r<8;++r) {
    int rr = r + 8*hi;
    Sp[(size_t)(n0+rr)*NN    + (m0+m)]    = a00[r];
    Sp[(size_t)(n0+rr)*NN    + (m0+16+m)] = a01[r];
    Sp[(size_t)(n0+16+rr)*NN + (m0+m)]    = a10[r];
    Sp[(size_t)(n0+16+rr)*NN + (m0+16+m)] = a11[r];
  }
}

// ---------------- row softmax (wave32, shuffle reduce) ---------------------------
__global__ __launch_bounds__(32) void softmax_rows(float* __restrict__ S)
{
  int row = blockIdx.x;                      // 12*512 rows
  float* p = S + (size_t)row*NN;
  int lane = threadIdx.x;
  float mx = -1e30f;
  for (int i=lane;i<NN;i+=32) mx = fmaxf(mx, p[i]);
  for (int off=16;off;off>>=1) mx = fmaxf(mx, __shfl_xor(mx, off, 32));
  float sum = 0.f;
  for (int i=lane;i<NN;i+=32){ float e=__expf(p[i]-mx); p[i]=e; sum+=e; }
  for (int off=16;off;off>>=1) sum += __shfl_xor(sum, off, 32);
  float inv = 1.f/sum;
  for (int i=lane;i<NN;i+=32) p[i]*=inv;
}

// ---------------- O = P * V (K=512, WMMA) ----------------------------------------
__global__ __launch_bounds__(32) void attn_o_wmma(const float* __restrict__ S,
                                                  const float* __restrict__ V,
                                                  float* __restrict__ O, int b)
{
  const int to = blockIdx.z;
  const int n0 = blockIdx.x*32, c0 = blockIdx.y*32;
  const float* Sp = S + (size_t)to*NN*NN;
  const float* Vp = V + (size_t)(b*TOUTP+to)*NN*TIN;
  const int lane = threadIdx.x, m = lane & 15, hi = lane >> 4;
  v8f a00={0,0,0,0,0,0,0,0}, a01=a00, a10=a00, a11=a00;
  #pragma unroll 8
  for (int k0=0;k0<NN;k0+=4) {
    int kl = k0 + 2*hi;
    v2f ar0 = { Sp[(size_t)(n0+m)*NN+kl],    Sp[(size_t)(n0+m)*NN+kl+1]    };
    v2f ar1 = { Sp[(size_t)(n0+16+m)*NN+kl], Sp[(size_t)(n0+16+m)*NN+kl+1] };
    v2f bc0 = { Vp[kl*TIN + c0+m],    Vp[(kl+1)*TIN + c0+m]    };
    v2f bc1 = { Vp[kl*TIN + c0+16+m], Vp[(kl+1)*TIN + c0+16+m] };
    a00 = wmma4(ar0,bc0,a00); a01 = wmma4(ar0,bc1,a01);
    a10 = wmma4(ar1,bc0,a10); a11 = wmma4(ar1,bc1,a11);
  }
  float* Op = O + (size_t)to*NN*TIN;
  #pragma unroll
  for (int r=0;r<8;++r) {
    int rr = r + 8*hi;
    Op[(size_t)(n0+rr)*TIN    + (c0+m)]    = a00[r];
    Op[(size_t)(n0+rr)*TIN    + (c0+16+m)] = a01[r];
    Op[(size_t)(n0+16+rr)*TIN + (c0+m)]    = a10[r];
    Op[(size_t)(n0+16+rr)*TIN + (c0+16+m)] = a11[r];
  }
}

// ---------------- head out: contract T' -> CL, write to concat buffer ------------
__global__ void head_out_kernel(const float* __restrict__ O, const float* __restrict__ ow,
                                const float* __restrict__ ob, float* __restrict__ CAT,
                                int b, int h, int Tp)
{
  size_t idx = (size_t)blockIdx.x*blockDim.x + threadIdx.x;
  if (idx >= (size_t)NN*CLH*TOUTP) return;
  int to = (int)(idx % TOUTP); size_t r = idx / TOUTP;
  int cl = (int)(r % CLH); int n = (int)(r / CLH);
  const float* op = O + ((size_t)to*NN+n)*TIN;
  float acc = ob[cl];
  for (int c=0;c<Tp;++c) acc += ow[cl*Tp+c]*op[c];
  CAT[(((size_t)b*NN+n)*(2*CLH) + h*CLH+cl)*TOUTP + to] = acc;
}

// ---------------- head residual: 1x1 conv 64x64 over h ---------------------------
__global__ void res_full_kernel(const float* __restrict__ HS, const float* __restrict__ rw,
                                const float* __restrict__ rb, float* __restrict__ RF, int b)
{
  size_t idx = (size_t)blockIdx.x*blockDim.x + threadIdx.x;
  if (idx >= (size_t)NN*CLH*TIN) return;
  int t = (int)(idx % TIN); size_t r = idx / TIN;
  int cl = (int)(r % CLH); int n = (int)(r / CLH);
  const float* hp = HS + (size_t)(b*NN+n)*SEQ64;
  float acc = rb[cl];
  for (int c2=0;c2<CLH;++c2) acc += rw[cl*CLH+c2]*hp[c2*TIN+t];
  RF[((size_t)n*CLH+cl)*TIN + t] = acc;
}

// ---------------- head residual time-contract + add + BN -------------------------
__global__ void head_bn_kernel(const float* __restrict__ RF, const float* __restrict__ rtw,
                               const float* __restrict__ rtb,
                               const float* __restrict__ g, const float* __restrict__ be,
                               const float* __restrict__ mean, const float* __restrict__ var,
                               float* __restrict__ CAT, int b, int h)
{
  size_t idx = (size_t)blockIdx.x*blockDim.x + threadIdx.x;
  if (idx >= (size_t)NN*CLH*TOUTP) return;
  int to = (int)(idx % TOUTP); size_t r = idx / TOUTP;
  int cl = (int)(r % CLH); int n = (int)(r / CLH);
  const float* rp = RF + ((size_t)n*CLH+cl)*TIN;
  float rr = rtb[to];
  for (int t=0;t<TIN;++t) rr += rtw[to*TIN+t]*rp[t];
  size_t ci = (((size_t)b*NN+n)*(2*CLH) + h*CLH+cl)*TOUTP + to;
  float v = CAT[ci] + rr;
  CAT[ci] = (v - mean[cl]) * (g[cl]*rsqrtf(var[cl]+EPSV)) + be[cl];
}

// ---------------- spatial model residual (time-contract of HS) -------------------
__global__ void spat_res_kernel(const float* __restrict__ HS, const float* __restrict__ rtw,
                                const float* __restrict__ rtb, float* __restrict__ SR)
{
  size_t idx = (size_t)blockIdx.x*blockDim.x + threadIdx.x;
  if (idx >= (size_t)NBN*CLH*TOUTP) return;
  int to = (int)(idx % TOUTP); size_t r = idx / TOUTP;
  int cl = (int)(r % CLH); int bn = (int)(r / CLH);
  const float* hp = HS + (size_t)bn*SEQ64 + cl*TIN;
  float acc = rtb[to];
  for (int t=0;t<TIN;++t) acc += rtw[to*TIN+t]*hp[t];
  SR[((size_t)bn*CLH+cl)*TOUTP+to] = acc;
}

// ---------------- spatial mix (64x128) + BN + final projection -------------------
__global__ void spat_final_kernel(const float* __restrict__ CAT, const float* __restrict__ SR,
                                  const float* __restrict__ ow, const float* __restrict__ ob,
                                  const float* __restrict__ g, const float* __restrict__ be,
                                  const float* __restrict__ mean, const float* __restrict__ var,
                                  const float* __restrict__ fw, const float* __restrict__ fb,
                                  float* __restrict__ SOUT)
{
  size_t idx = (size_t)blockIdx.x*blockDim.x + threadIdx.x;
  if (idx >= (size_t)NB*TOUTP*NN) return;
  int n = (int)(idx % NN); size_t r = idx / NN;
  int p = (int)(r % TOUTP); int b = (int)(r / TOUTP);
  size_t bn = (size_t)b*NN + n;
  const float* cp = CAT + bn*(2*CLH)*TOUTP;
  float acc = fb[0];
  for (int cl=0;cl<CLH;++cl) {
    float u = ob[cl];
    for (int hc=0;hc<2*CLH;++hc) u += ow[cl*(2*CLH)+hc]*cp[hc*TOUTP+p];
    u += SR[(bn*CLH+cl)*TOUTP+p];
    u = (u - mean[cl]) * (g[cl]*rsqrtf(var[cl]+EPSV)) + be[cl];
    acc += fw[cl]*u;
  }
  SOUT[((size_t)b*TOUTP+p)*NN+n] = acc;
}

// ---------------- final 2-way blend ----------------------------------------------
__global__ void final_kernel(const float* __restrict__ TO, const float* __restrict__ SO,
                             const float* __restrict__ w2, const float* __restrict__ b1,
                             float* __restrict__ out)
{
  size_t idx = (size_t)blockIdx.x*blockDim.x + threadIdx.x;
  if (idx >= (size_t)NB*TOUTP*NN) return;
  out[idx] = b1[0] + w2[0]*TO[idx] + w2[1]*SO[idx];
}

// =================================================================================
extern "C" void kernel_launch(void* const* d_in, const int* in_sizes, int n_in,
                              void* d_out, int out_size, void* d_ws, size_t ws_size,
                              hipStream_t stream)
{
  (void)in_sizes; (void)out_size;
  if (n_in < 128 || !d_out || !d_ws) return;
  #define F(i) ((const float*)d_in[(i)])

  // ---- workspace layout (floats) ----
  const size_t BIGT   = (size_t)NBN*SEQ32;                 // 12,582,912
  const size_t OUTSSZ = (size_t)NBN*RCH*TOUTP;             //  1,572,864 per stack
  float* ws    = (float*)d_ws;
  float* RES_A = ws + 0*BIGT;
  float* RES_B = ws + 1*BIGT;
  float* GF    = ws + 2*BIGT;
  float* SKA   = ws + 3*BIGT;
  float* SKB   = ws + 4*BIGT;
  float* OUTS  = ws + 5*BIGT;
  float* TOUTb = OUTS + 2*OUTSSZ;                          // [b][p][n]
  float* ATTN  = TOUTb + (size_t)NB*TOUTP*NN;              // per-b: 12*512*512
  float* RF    = ATTN  + (size_t)TOUTP*NN*NN;              // per-b: 512*64*96
  float* CAT   = RF    + (size_t)NN*CLH*TIN;               // [b][n][128][12]
  float* SRES  = CAT   + (size_t)NBN*2*CLH*TOUTP;          // [b][n][64][12]
  float* SOUT  = SRES  + (size_t)NBN*CLH*TOUTP;            // [b][p][n]
  const size_t TOTALF = (size_t)(SOUT + (size_t)NB*TOUTP*NN - ws);
  if (ws_size < TOTALF*sizeof(float)) return;

  // ---- parameter unpack (setup_inputs insertion order) ----
  const float* hist = F(0);                                // d_in[1]=fut_ts unused by model
  int pi = 2;
  const float* t_in_w = F(pi++); const float* t_in_b = F(pi++);
  const float *fw[2][4], *fb[2][4], *gw[2][4], *gb[2][4], *sw_[2][4], *sb_[2][4],
              *rw_[2][4], *rb_[2][4], *tw_[2][4], *tb_[2][4], *lw[2], *lb[2];
  const int Ls[2] = {4,3}; const int ks_[2] = {2,3};
  for (int j=0;j<2;++j) { int L=Ls[j];
    for (int i=0;i<L;++i) fw[j][i]=F(pi++);  for (int i=0;i<L;++i) fb[j][i]=F(pi++);
    for (int i=0;i<L;++i) gw[j][i]=F(pi++);  for (int i=0;i<L;++i) gb[j][i]=F(pi++);
    for (int i=0;i<L;++i) sw_[j][i]=F(pi++); for (int i=0;i<L;++i) sb_[j][i]=F(pi++);
    for (int i=0;i<L;++i) rw_[j][i]=F(pi++); for (int i=0;i<L;++i) rb_[j][i]=F(pi++);
    for (int i=0;i<L;++i) tw_[j][i]=F(pi++); for (int i=0;i<L;++i) tb_[j][i]=F(pi++);
    lw[j]=F(pi++); lb[j]=F(pi++);
  }
  const float* t_stack_w=F(pi++); const float* t_stack_b=F(pi++);
  const float* t_final_w=F(pi++); const float* t_final_b=F(pi++);
  const float* s_in_w=F(pi++);    const float* s_in_b=F(pi++);
  const float *hkw[2],*hkb[2],*hqw[2],*hqb[2],*hvw[2],*hvb[2],*how_[2],*hob[2],
              *hrw[2],*hrb[2],*hrtw[2],*hrtb[2],*hbg[2],*hbb[2],*hbm[2],*hbv[2];
  for (int h=0;h<2;++h) {
    hkw[h]=F(pi++); hkb[h]=F(pi++); hqw[h]=F(pi++); hqb[h]=F(pi++);
    hvw[h]=F(pi++); hvb[h]=F(pi++); how_[h]=F(pi++); hob[h]=F(pi++);
    hrw[h]=F(pi++); hrb[h]=F(pi++); hrtw[h]=F(pi++); hrtb[h]=F(pi++);
    hbg[h]=F(pi++); hbb[h]=F(pi++); hbm[h]=F(pi++); hbv[h]=F(pi++);
  }
  const float* s_rtw=F(pi++); const float* s_rtb=F(pi++);
  const float* s_ow =F(pi++); const float* s_ob =F(pi++);
  const float* s_bg =F(pi++); const float* s_bb =F(pi++);
  const float* s_bm =F(pi++); const float* s_bv =F(pi++);
  const float* s_fw =F(pi++); const float* s_fb =F(pi++);
  const float* g_fw =F(pi++); const float* g_fb =F(pi++);

  auto cdiv = [](size_t a, size_t b)->unsigned { return (unsigned)((a + b - 1)/b); };

  // ================= TEMPORAL MODEL =================
  for (int j=0;j<2;++j) {
    const int L = Ls[j], kk = ks_[j];
    in_conv_kernel<<<cdiv((size_t)NBN*TIN,256),256,0,stream>>>(hist, t_in_w, t_in_b, RES_A, RCH);
    float *rA=RES_A, *rB=RES_B, *sA=SKA, *sB=SKB;
    int Lcur=TIN, skipLen=0, first=1;
    for (int i=0;i<L;++i) {
      const int d = 1<<(i+1);
      const int Lout = Lcur - d*(kk-1);
      if (kk == 2)
        gated_conv_wmma<2><<<NBN,32,0,stream>>>(rA, GF, fw[j][i], fb[j][i], gw[j][i], gb[j][i], Lcur, d);
      else
        gated_conv_wmma<3><<<NBN,32,0,stream>>>(rA, GF, fw[j][i], fb[j][i], gw[j][i], gb[j][i], Lcur, d);
      pw32_shift_add<<<cdiv((size_t)NBN*RCH*Lout,256),256,0,stream>>>(
          GF, sA, sB, sw_[j][i], sb_[j][i], Lout, first?0:(skipLen-Lout), first?0:1);
      { float* t=sA; sA=sB; sB=t; } skipLen=Lout; first=0;
      pw32_shift_add<<<cdiv((size_t)NBN*RCH*Lout,256),256,0,stream>>>(
          GF, rA, rB, rw_[j][i], rb_[j][i], Lout, d*(kk-1), 1);
      { float* t=rA; rA=rB; rB=t; } Lcur=Lout;
    }
    for (int i=L-1;i>=0;--i) {
      const int d = 1<<(i+1);
      tconv_kernel<<<cdiv((size_t)NBN*RCH*(skipLen+d*(kk-1)),256),256,0,stream>>>(
          sA, sB, tw_[j][i], tb_[j][i], skipLen, d, kk);
      { float* t=sA; sA=sB; sB=t; } skipLen += d*(kk-1);
    }
    lin_time_kernel<<<cdiv((size_t)NBN*RCH*TOUTP,256),256,0,stream>>>(sA, OUTS + (size_t)j*OUTSSZ, lw[j], lb[j]);
  }
  temp_final_kernel<<<cdiv((size_t)NB*TOUTP*NN,256),256,0,stream>>>(
      OUTS, TOUTb, t_stack_w, t_stack_b, t_final_w, t_final_b);

  // ================= SPATIAL MODEL (reuses temporal buffers) =================
  float* HS = RES_A;                                       // 25.17M floats: RES_A+RES_B
  float* KK = GF;
  float* QQ = KK + (size_t)NB*TOUTP*NN*TIN;
  float* VV = QQ + (size_t)NB*TOUTP*NN*TIN;                // spans GF+SKA regions
  float* OO = SKB;                                         // per-b: 12*512*96

  in_conv_kernel<<<cdiv((size_t)NBN*TIN,256),256,0,stream>>>(hist, s_in_w, s_in_b, HS, CLH);

  for (int h=0;h<2;++h) {
    const int ksz = ks_[h], Tp = TIN - ksz + 1;
    kqv_kernel<<<cdiv((size_t)NBN*TIN,256),256,0,stream>>>(
        HS, hkw[h],hkb[h], hqw[h],hqb[h], hvw[h],hvb[h], KK,QQ,VV, ksz);
    for (int b=0;b<NB;++b) {
      attn_s_wmma<<<dim3(16,16,TOUTP),32,0,stream>>>(KK, QQ, ATTN, b);
      softmax_rows<<<TOUTP*NN,32,0,stream>>>(ATTN);
      attn_o_wmma<<<dim3(16,3,TOUTP),32,0,stream>>>(ATTN, VV, OO, b);
      head_out_kernel<<<cdiv((size_t)NN*CLH*TOUTP,256),256,0,stream>>>(OO, how_[h], hob[h], CAT, b, h, Tp);
      res_full_kernel<<<cdiv((size_t)NN*CLH*TIN,256),256,0,stream>>>(HS, hrw[h], hrb[h], RF, b);
      head_bn_kernel<<<cdiv((size_t)NN*CLH*TOUTP,256),256,0,stream>>>(
          RF, hrtw[h], hrtb[h], hbg[h],hbb[h],hbm[h],hbv[h], CAT, b, h);
    }
  }
  spat_res_kernel<<<cdiv((size_t)NBN*CLH*TOUTP,256),256,0,stream>>>(HS, s_rtw, s_rtb, SRES);
  spat_final_kernel<<<cdiv((size_t)NB*TOUTP*NN,256),256,0,stream>>>(
      CAT, SRES, s_ow, s_ob, s_bg, s_bb, s_bm, s_bv, s_fw, s_fb, SOUT);
  final_kernel<<<cdiv((size_t)NB*TOUTP*NN,256),256,0,stream>>>(TOUTb, SOUT, g_fw, g_fb, (float*)d_out);
  #undef F
}